// GCNBlock_48747878809894
// MI455X (gfx1250) — compile-verified
//
#include <hip/hip_runtime.h>
#include <math.h>

typedef float v2f __attribute__((ext_vector_type(2)));
typedef float v8f __attribute__((ext_vector_type(8)));

#define DFEAT 64
#define PAIR_STRIDE 160   // 128 data floats + 32 pad floats -> adjacent pair-rows
                          // land on disjoint 32-bank halves (conflict-free b64 reads)

// ---------------- degree / normalization ----------------

__global__ void gcn_init_deg(float* __restrict__ deg, int n) {
    int i = blockIdx.x * blockDim.x + threadIdx.x;
    if (i < n) deg[i] = 1.0f;                      // self-loop weight
}

__global__ void gcn_scatter_deg(const int* __restrict__ dst,
                                const float* __restrict__ ew,
                                float* __restrict__ deg, int E) {
    int e = blockIdx.x * blockDim.x + threadIdx.x;
    if (e < E) atomicAdd(&deg[dst[e]], ew[e]);
}

__global__ void gcn_rsqrt_deg(float* __restrict__ deg, int n) {
    int i = blockIdx.x * blockDim.x + threadIdx.x;
    if (i < n) {
        float d = deg[i];
        deg[i] = (d > 0.0f) ? rsqrtf(d) : 0.0f;    // becomes dinv[]
    }
}

// ---------------- FP32 WMMA GEMM: out = A[n x 64] * W[64 x 64] ----------------
// One wave computes a 16(M) x 64(N) strip: 4 accumulators, K stepped by 4
// using V_WMMA_F32_16X16X4_F32.
// A fragment (ISA 7.12.2, 32-bit A 16x4): lanes 0-15 hold K={0,1}; 16-31 K={2,3}.
// W is staged in LDS pair-interleaved: for K-pair p=(k>>1),
//   ldsW[p*PAIR_STRIDE + 2*n + (k&1)] = W[k][n]
// so each B fragment is ONE aligned ds_load_b64 straight into an even VGPR pair.

__global__ void __launch_bounds__(256)
gcn_gemm_wmma(const float* __restrict__ A, const float* __restrict__ W,
              float* __restrict__ out, int nRows) {
    __shared__ float ldsW[(DFEAT / 2) * PAIR_STRIDE];   // 32 * 160 floats = 20 KB
    for (int i = threadIdx.x; i < DFEAT * DFEAT; i += 256) {
        int k = i >> 6;        // 0..63
        int n = i & 63;
        ldsW[(k >> 1) * PAIR_STRIDE + 2 * n + (k & 1)] = W[i];
    }
    __syncthreads();

    int wave = blockIdx.x * (blockDim.x >> 5) + (threadIdx.x >> 5);
    int lane = threadIdx.x & 31;
    int m0 = wave << 4;
    if (m0 >= nRows) return;

    int half = lane >> 4;          // 0: K pair {0,1}, 1: K pair {2,3}
    int sub  = lane & 15;          // M row (for A) / N col (for B)
    int row  = m0 + sub;
    if (row >= nRows) row = nRows - 1;   // clamp loads; stores are guarded
    const float* arow = A + (long)row * DFEAT;

    v8f acc0 = {}; v8f acc1 = {}; v8f acc2 = {}; v8f acc3 = {};

    #pragma unroll
    for (int k = 0; k < DFEAT; k += 4) {
        v2f a = *(const v2f*)(arow + k + 2 * half);   // 8B aligned (k even)
        // pair-row index for this half: p = k/2 + half
        const float* wp = &ldsW[((k >> 1) + half) * PAIR_STRIDE + 2 * sub];
        v2f b0 = *(const v2f*)(wp);          // N tile 0: cols sub+0
        v2f b1 = *(const v2f*)(wp + 32);     // N tile 1: cols sub+16
        v2f b2 = *(const v2f*)(wp + 64);     // N tile 2: cols sub+32
        v2f b3 = *(const v2f*)(wp + 96);     // N tile 3: cols sub+48
        acc0 = __builtin_amdgcn_wmma_f32_16x16x4_f32(false, a, false, b0, (short)0, acc0, false, false);
        acc1 = __builtin_amdgcn_wmma_f32_16x16x4_f32(false, a, false, b1, (short)0, acc1, false, false);
        acc2 = __builtin_amdgcn_wmma_f32_16x16x4_f32(false, a, false, b2, (short)0, acc2, false, false);
        acc3 = __builtin_amdgcn_wmma_f32_16x16x4_f32(false, a, false, b3, (short)0, acc3, false, false);
    }

    // C/D layout: VGPR v -> M = v + 8*half, N = lane%16 (+16*tile)
    #pragma unroll
    for (int v = 0; v < 8; ++v) {
        int r = m0 + v + 8 * half;
        if (r < nRows) {
            float* orow = out + (long)r * DFEAT + sub;
            orow[0]  = acc0[v];
            orow[16] = acc1[v];
            orow[32] = acc2[v];
            orow[48] = acc3[v];
        }
    }
}

// ---------------- aggregation ----------------

__global__ void gcn_self_init(const float* __restrict__ hw,
                              const float* __restrict__ dinv,
                              float* __restrict__ agg, long total) {
    long t = (long)blockIdx.x * blockDim.x + threadIdx.x;
    if (t < total) {
        float di = dinv[t >> 6];
        agg[t] = hw[t] * di * di;                  // self-loop: norm = dinv[i]^2
    }
}

__global__ void gcn_edge_scatter(const int* __restrict__ src,
                                 const int* __restrict__ dst,
                                 const float* __restrict__ ew,
                                 const float* __restrict__ dinv,
                                 const float* __restrict__ hw,
                                 float* __restrict__ agg, int E) {
    long t = (long)blockIdx.x * blockDim.x + threadIdx.x;
    int e = (int)(t >> 6);
    int c = (int)(t & 63);
    if (e >= E) return;
    int s = src[e];
    int d = dst[e];
    float nrm = dinv[s] * ew[e] * dinv[d];
    atomicAdd(&agg[(long)d * DFEAT + c], hw[(long)s * DFEAT + c] * nrm);
}

__global__ void gcn_bias_gelu(const float* __restrict__ agg,
                              const float* __restrict__ bias,
                              float* __restrict__ h, long total) {
    long t = (long)blockIdx.x * blockDim.x + threadIdx.x;
    if (t < total) {
        float x = agg[t] + bias[t & 63];
        h[t] = 0.5f * x * (1.0f + erff(x * 0.70710678118654752f));  // exact GELU
    }
}

// ---------------- mean pool (deterministic; batch is sorted) ----------------

__global__ void __launch_bounds__(256)
gcn_mean_pool(const float* __restrict__ h, const int* __restrict__ batch,
              float* __restrict__ outp, int n) {
    int g = blockIdx.x;
    // lower_bound(batch, g) and lower_bound(batch, g+1)
    int lo = 0, hi = n;
    while (lo < hi) { int mid = (lo + hi) >> 1; if (batch[mid] < g) lo = mid + 1; else hi = mid; }
    int start = lo;
    hi = n;
    while (lo < hi) { int mid = (lo + hi) >> 1; if (batch[mid] < g + 1) lo = mid + 1; else hi = mid; }
    int end = lo;

    int col = threadIdx.x & 63;
    int sub = threadIdx.x >> 6;    // 0..3
    float sum = 0.0f;
    for (int r = start + sub; r < end; r += 4) sum += h[(long)r * DFEAT + col];

    __shared__ float red[256];
    red[threadIdx.x] = sum;
    __syncthreads();
    if (threadIdx.x < 64) {
        float s = red[threadIdx.x] + red[threadIdx.x + 64] +
                  red[threadIdx.x + 128] + red[threadIdx.x + 192];
        float cnt = (float)(end - start);
        outp[(long)g * DFEAT + threadIdx.x] = s / fmaxf(cnt, 1.0f);
    }
}

// ---------------- launcher ----------------

extern "C" void kernel_launch(void* const* d_in, const int* in_sizes, int n_in,
                              void* d_out, int out_size, void* d_ws, size_t ws_size,
                              hipStream_t stream) {
    const float* x     = (const float*)d_in[0];
    const int*   ei    = (const int*)d_in[1];   // [2, E] flat: row0 = src, row1 = dst
    const float* ew    = (const float*)d_in[2];
    const int*   batch = (const int*)d_in[3];
    const float* W0    = (const float*)d_in[4];
    const float* b0    = (const float*)d_in[5];
    const float* W1    = (const float*)d_in[6];
    const float* b1    = (const float*)d_in[7];

    const int N = in_sizes[0] / DFEAT;
    const int E = in_sizes[1] / 2;
    const int G = out_size / (2 * DFEAT);
    const int* src = ei;
    const int* dst = ei + E;

    float* ws   = (float*)d_ws;
    float* dinv = ws;                       // N floats (deg -> dinv in place)
    float* hw   = dinv + N;                 // N*64
    float* agg  = hw + (long)N * DFEAT;     // N*64
    float* h    = agg + (long)N * DFEAT;    // N*64
    float* outp = (float*)d_out;

    const long total = (long)N * DFEAT;
    const int  TB = 256;
    const int  gridN   = (N + TB - 1) / TB;
    const int  gridE   = (E + TB - 1) / TB;
    const int  gridNE  = (int)((total + TB - 1) / TB);
    const int  gridE64 = (int)(((long)E * DFEAT + TB - 1) / TB);
    const int  waves   = (N + 15) / 16;
    const int  gridGemm = (waves + 7) / 8;  // 8 waves / 256-thread block

    // normalization: deg = 1 + sum_{e: dst==i} ew[e]; dinv = rsqrt(deg)
    gcn_init_deg<<<gridN, TB, 0, stream>>>(dinv, N);
    gcn_scatter_deg<<<gridE, TB, 0, stream>>>(dst, ew, dinv, E);
    gcn_rsqrt_deg<<<gridN, TB, 0, stream>>>(dinv, N);

    const float* hin = x;
    const float* Ws[2] = {W0, W1};
    const float* bs[2] = {b0, b1};
    for (int layer = 0; layer < 2; ++layer) {
        gcn_gemm_wmma<<<gridGemm, TB, 0, stream>>>(hin, Ws[layer], hw, N);
        gcn_self_init<<<gridNE, TB, 0, stream>>>(hw, dinv, agg, total);
        gcn_edge_scatter<<<gridE64, TB, 0, stream>>>(src, dst, ew, dinv, hw, agg, E);
        gcn_bias_gelu<<<gridNE, TB, 0, stream>>>(agg, bs[layer], h, total);
        gcn_mean_pool<<<G, TB, 0, stream>>>(h, batch, outp + (long)layer * G * DFEAT, N);
        hin = h;  // layer 1 consumes post-GELU features of layer 0
    }
}